// CustomerLinear_89644557402993
// MI455X (gfx1250) — compile-verified
//
#include <hip/hip_runtime.h>
#include <cstdint>

typedef __attribute__((ext_vector_type(4))) float v4f;

#define HDIM 4096
#define GROUP 128
#define NGROUPS (HDIM / GROUP)            // 32 groups per token
#define BLOCK 256                          // 8 wave32s
#define EPT (HDIM / BLOCK)                 // 16 elements per thread
#define FP8_MAXV 448.0f
#define SCALE_MIN 1e-10f
#define RMS_EPS 1e-6f

__global__ __launch_bounds__(BLOCK) void fused_addrms_fp8q(
    const float* __restrict__ x,
    const float* __restrict__ residual,
    const float* __restrict__ weight,
    float* __restrict__ out_q,      // [T,H] fp8 values decoded to f32
    float* __restrict__ out_scale,  // [T, H/128]
    float* __restrict__ out_res)    // [T,H]
{
    __shared__ float s_w[HDIM];           // 16 KB weight tile (async DMA target)
    __shared__ float s_sum[BLOCK / 32];   // per-wave partial sums

    const int tid = threadIdx.x;
    const long long row = blockIdx.x;
    const long long base = row * (long long)HDIM + (long long)tid * EPT;

    // --- CDNA5 async DMA: pull the first 4KB (1024 floats) of the weight vector into LDS,
    // overlapped with the streaming loads below. 256 threads x b128 = 4KB.
    {
        unsigned lds_off = (unsigned)(uintptr_t)(&s_w[tid * 4]);
        const float* gptr = weight + tid * 4;
        asm volatile("global_load_async_to_lds_b128 %0, %1, off"
                     :: "v"(lds_off), "v"(gptr) : "memory");
    }

    // --- Streaming load x/residual (non-temporal: 1GB stream >> 192MB L2)
    v4f vx[4], vr[4], vv[4];
    const v4f* xp = (const v4f*)(x + base);
    const v4f* rp = (const v4f*)(residual + base);
#pragma unroll
    for (int i = 0; i < 4; ++i) {
        vx[i] = __builtin_nontemporal_load(xp + i);
        vr[i] = __builtin_nontemporal_load(rp + i);
    }

    float ss = 0.0f;
#pragma unroll
    for (int i = 0; i < 4; ++i) {
        vv[i] = vx[i] + vr[i];
#pragma unroll
        for (int j = 0; j < 4; ++j) ss += vv[i][j] * vv[i][j];
    }

    // residual output (streaming store)
    v4f* resp = (v4f*)(out_res + base);
#pragma unroll
    for (int i = 0; i < 4; ++i) __builtin_nontemporal_store(vv[i], resp + i);

    // --- wave32 tree reduction of sum-of-squares
#pragma unroll
    for (int d = 1; d < 32; d <<= 1) ss += __shfl_xor(ss, d, 32);
    if ((tid & 31) == 0) s_sum[tid >> 5] = ss;

    // weight DMA must have landed before anyone reads s_w after the barrier
    asm volatile("s_wait_asynccnt 0x0" ::: "memory");
    __syncthreads();

    float tot = 0.0f;
#pragma unroll
    for (int w = 0; w < BLOCK / 32; ++w) tot += s_sum[w];
    const float inv_rms = rsqrtf(tot * (1.0f / (float)HDIM) + RMS_EPS);

    // --- y = res * inv_rms * w ; per-thread amax over its 16 elements
    float y[EPT];
    float amax = 0.0f;
#pragma unroll
    for (int i = 0; i < 4; ++i) {
#pragma unroll
        for (int j = 0; j < 4; ++j) {
            const float wv = s_w[tid * EPT + i * 4 + j];
            const float yy = vv[i][j] * inv_rms * wv;
            y[i * 4 + j] = yy;
            amax = fmaxf(amax, fabsf(yy));
        }
    }

    // group of 128 elems = 8 consecutive lanes -> 3-step xor-shuffle amax
#pragma unroll
    for (int d = 1; d < 8; d <<= 1) amax = fmaxf(amax, __shfl_xor(amax, d, 32));

    const float scale = fmaxf(amax, SCALE_MIN) * (1.0f / FP8_MAXV);
    const float rscale = 1.0f / scale;
    if ((tid & 7) == 0)
        out_scale[row * NGROUPS + (tid >> 3)] = scale;

    // --- quantize: clamp, HW convert to fp8 e4m3 (RNE), decode back to f32
    v4f qo[4];
#pragma unroll
    for (int i = 0; i < 4; ++i) {
#pragma unroll
        for (int p = 0; p < 2; ++p) {
            float q0 = fminf(fmaxf(y[i * 4 + p * 2 + 0] * rscale, -FP8_MAXV), FP8_MAXV);
            float q1 = fminf(fmaxf(y[i * 4 + p * 2 + 1] * rscale, -FP8_MAXV), FP8_MAXV);
            int packed = __builtin_amdgcn_cvt_pk_fp8_f32(q0, q1, 0, false);
            qo[i][p * 2 + 0] = __builtin_amdgcn_cvt_f32_fp8(packed, 0);
            qo[i][p * 2 + 1] = __builtin_amdgcn_cvt_f32_fp8(packed, 1);
        }
    }
    v4f* qp = (v4f*)(out_q + base);
#pragma unroll
    for (int i = 0; i < 4; ++i) __builtin_nontemporal_store(qo[i], qp + i);
}

extern "C" void kernel_launch(void* const* d_in, const int* in_sizes, int n_in,
                              void* d_out, int out_size, void* d_ws, size_t ws_size,
                              hipStream_t stream) {
    const float* x        = (const float*)d_in[0];
    const float* residual = (const float*)d_in[1];
    const float* weight   = (const float*)d_in[2];

    const long long T = (long long)in_sizes[0] / HDIM;   // 16384

    // d_out layout = reference tuple flattened: q [T,H] | scales [T,32] | res [T,H]
    float* out_q     = (float*)d_out;
    float* out_scale = out_q + T * (long long)HDIM;
    float* out_res   = out_scale + T * (long long)NGROUPS;

    fused_addrms_fp8q<<<dim3((unsigned)T), dim3(BLOCK), 0, stream>>>(
        x, residual, weight, out_q, out_scale, out_res);
}